// Cochlea_32667521253868
// MI455X (gfx1250) — compile-verified
//
#include <hip/hip_runtime.h>

typedef __attribute__((ext_vector_type(2))) float v2f;
typedef __attribute__((ext_vector_type(8))) float v8f;

#define ALPHA   0.9f
#define ONE_MA  0.1f
#define T_DIM   4096
#define F_DIM   64
#define OUT_T   512   // output timesteps per block
#define WARM_T  256   // warm-up prefix (0.9^256 ~ 2e-12 -> negligible in fp32)

// One wave (32 lanes) per block. Each block owns a 16-chain column tile and a
// 512-step T chunk (plus 256-step warm-up). The IIR scan over each 16-step
// segment is computed as Y = L*X + d*carry with four chained
// v_wmma_f32_16x16x4_f32 ops (K = 16 total).
__global__ __launch_bounds__(32) void cochlea_scan_wmma(
    const float* __restrict__ spec,   // (32, 4096, 64)
    const float* __restrict__ ftune,  // (64,)
    const float* __restrict__ thr_p,  // scalar
    const float* __restrict__ ratio_p,// scalar
    float* __restrict__ out)          // (32, 4096, 64)
{
    const int lane = threadIdx.x;
    const int n    = lane & 15;   // tile column (chain) for B/C/D; M row for A
    const int half = lane >> 4;   // K/M split between lane halves

    const int bid   = blockIdx.x;   // 0..1023
    const int ct    = bid >> 3;     // column tile 0..127
    const int chunk = bid & 7;      // T chunk 0..7
    const int b     = ct >> 2;      // batch
    const int f0    = (ct & 3) << 4;
    const int f     = f0 + n;

    const float thr   = thr_p[0];
    const float ratio = ratio_p[0];
    const float tune  = ftune[f];

    // Powers of alpha, iteratively (matches reference's repeated-multiply style)
    float apow[17];
    apow[0] = 1.0f;
#pragma unroll
    for (int i = 1; i <= 16; ++i) apow[i] = apow[i - 1] * ALPHA;

    // A operands: L[M, s] = 0.1 * alpha^(M-s) for s <= M, else 0.
    // A 16x4 f32 layout: lane M = lane&15; VGPR j holds K = j + 2*half.
    v2f Aop[4];
#pragma unroll
    for (int kb = 0; kb < 4; ++kb) {
#pragma unroll
        for (int j = 0; j < 2; ++j) {
            const int s = 4 * kb + 2 * half + j;
            Aop[kb][j] = (n >= s) ? (ONE_MA * apow[n - s]) : 0.0f;
        }
    }

    // Carry coefficients for C init: d[m] = alpha^(M+1), M = m + 8*half
    float dcoef[8];
#pragma unroll
    for (int m = 0; m < 8; ++m) dcoef[m] = apow[m + 8 * half + 1];

    const int t0 = chunk * OUT_T;
    const int tw = (chunk == 0) ? 0 : (t0 - WARM_T);
    const int warmSegs = (t0 - tw) >> 4;
    const int nSeg     = warmSegs + (OUT_T >> 4);

    const float* sp = spec + b * (T_DIM * F_DIM);
    float*       op = out  + b * (T_DIM * F_DIM);

    // Initial carry: exact boundary handling at t==0 (carry = compressed x[0]
    // makes the uniform L formula reproduce y[0] = x[0]); zero otherwise.
    float carry = 0.0f;
    if (chunk == 0) {
        float x0 = sp[f] * tune;
        carry = (x0 > thr) ? (thr + (x0 - thr) * ratio) : x0;
    }

    for (int seg = 0; seg < nSeg; ++seg) {
        const int tb = tw + (seg << 4);

        // Load + compress the 16x16 X tile.
        // B 4x16 f32 layout (mirror of A): lane N = lane&15; VGPR j -> K = j + 2*half.
        v2f Bop[4];
#pragma unroll
        for (int kb = 0; kb < 4; ++kb) {
#pragma unroll
            for (int j = 0; j < 2; ++j) {
                const int t = tb + 4 * kb + 2 * half + j;
                float x = sp[t * F_DIM + f] * tune;
                x = (x > thr) ? (thr + (x - thr) * ratio) : x;
                Bop[kb][j] = x;
            }
        }

        // C init with rank-1 carry term: C[M,n] = alpha^(M+1) * carry[n]
        v8f c;
#pragma unroll
        for (int m = 0; m < 8; ++m) c[m] = dcoef[m] * carry;

        // K = 16 via 4 chained 16x16x4 f32 WMMAs (accumulation hazard is
        // handled by the compiler-inserted waits).
#pragma unroll
        for (int kb = 0; kb < 4; ++kb) {
            c = __builtin_amdgcn_wmma_f32_16x16x4_f32(
                    false, Aop[kb], false, Bop[kb],
                    (short)0, c, false, false);
        }

        // New carry = y[M=15, n] = VGPR7 of upper-half lanes; broadcast per column.
        carry = __shfl(c[7], 16 + n, 32);

        // Store (ReLU) — D layout: lane n, VGPR m -> (t = tb + m + 8*half, chain n).
        if (seg >= warmSegs) {
#pragma unroll
            for (int m = 0; m < 8; ++m) {
                const int t = tb + m + 8 * half;
                op[t * F_DIM + f] = fmaxf(c[m], 0.0f);
            }
        }
    }
}

extern "C" void kernel_launch(void* const* d_in, const int* in_sizes, int n_in,
                              void* d_out, int out_size, void* d_ws, size_t ws_size,
                              hipStream_t stream) {
    const float* spec  = (const float*)d_in[0];
    const float* ftune = (const float*)d_in[1];
    const float* thr   = (const float*)d_in[2];
    const float* ratio = (const float*)d_in[3];
    float* out = (float*)d_out;

    dim3 grid(1024);   // 128 column tiles x 8 T-chunks
    dim3 block(32);    // one wave32 per block
    hipLaunchKernelGGL(cochlea_scan_wmma, grid, block, 0, stream,
                       spec, ftune, thr, ratio, out);
}